// Llama4TextMoe_77034533421086
// MI455X (gfx1250) — compile-verified
//
#include <hip/hip_runtime.h>

// ---------------------------------------------------------------------------
// Llama4 MoE (B=1,S=2048,H=1024,I=4096,E=8,TOP_K=1), fp32, MI455X/gfx1250.
// Memory-bound: ~0.55 GB weights+activations @ 23.3 TB/s ≈ 24 us, while the
// 103 GFLOP costs <2 us even on the fp32 V_WMMA_F32_16X16X4_F32 path — so we
// keep exact fp32 WMMA (matches the fp32 reference numerically).
// ---------------------------------------------------------------------------

typedef __attribute__((ext_vector_type(2))) float v2f;
typedef __attribute__((ext_vector_type(8))) float v8f;

constexpr int kT  = 2048;  // tokens
constexpr int kH  = 1024;  // hidden
constexpr int kI  = 4096;  // intermediate
constexpr int kE  = 8;     // experts
constexpr int kMT = 64;    // M (token) tile per block
constexpr int kKC = 32;    // K chunk staged in LDS
constexpr int kLDA = 33;   // padded LDS row stride (bank-conflict avoidance)

__device__ __forceinline__ float silu_f(float x) {
    return x / (1.0f + __expf(-x));
}

__device__ __forceinline__ v8f wmma_f32(v2f a, v2f b, v8f c) {
    return __builtin_amdgcn_wmma_f32_16x16x4_f32(
        false, a, false, b, (short)0, c, false, false);
}

// --------------------------- init: zero counters ---------------------------
__global__ void init_kernel(int* __restrict__ cnt) {
    if (threadIdx.x < kE) cnt[threadIdx.x] = 0;
}

// --------------------------- router: top-1 + lists -------------------------
// One wave32 per token. 8 waves per 256-thread block.
__global__ void router_kernel(const float* __restrict__ hs,
                              const float* __restrict__ rw,
                              float* __restrict__ scores_out,  // [E, T]
                              float* __restrict__ sc,          // [T]
                              int* __restrict__ cnt,           // [E]
                              int* __restrict__ list) {        // [E, T]
    const int wave = threadIdx.x >> 5;
    const int lane = threadIdx.x & 31;
    const int t = blockIdx.x * 8 + wave;
    if (t >= kT) return;

    float acc[kE];
#pragma unroll
    for (int e = 0; e < kE; ++e) acc[e] = 0.0f;

    for (int h = lane; h < kH; h += 32) {
        const float x = hs[(size_t)t * kH + h];
#pragma unroll
        for (int e = 0; e < kE; ++e) acc[e] += x * rw[(size_t)e * kH + h];
    }
#pragma unroll
    for (int e = 0; e < kE; ++e) {
#pragma unroll
        for (int off = 16; off > 0; off >>= 1)
            acc[e] += __shfl_xor(acc[e], off, 32);
    }
    if (lane == 0) {
        int best = 0;
        float bv = acc[0];
#pragma unroll
        for (int e = 1; e < kE; ++e)
            if (acc[e] > bv) { bv = acc[e]; best = e; }   // ties -> lowest idx
        const float s = 1.0f / (1.0f + __expf(-bv));
#pragma unroll
        for (int e = 0; e < kE; ++e)
            scores_out[(size_t)e * kT + t] = (e == best) ? s : 0.0f;
        sc[t] = s;
        const int pos = atomicAdd(&cnt[best], 1);
        list[best * kT + pos] = t;
    }
}

// ------------------- GEMM1: P = silu(X Wg^T) * (X Wu^T) --------------------
// Block = 128 threads (4 waves). Output tile: 64 tokens x 64 I-cols for both
// gate and up. Wave w owns 16 columns; 4 M-subtiles each (v8f accum).
__global__ __launch_bounds__(128) void gemm1_kernel(
    const float* __restrict__ X,        // [T, H]
    const float* __restrict__ Wg_base,  // routed: [E,I,H]; shared: [I,H]
    const float* __restrict__ Wu_base,
    float* __restrict__ P,              // [T, I]
    const int* __restrict__ cnt, const int* __restrict__ list,
    const float* __restrict__ sc, int routed) {
    __shared__ float Alds[kMT * kLDA];
    __shared__ int   tokLds[kMT];
    __shared__ float sclLds[kMT];

    const int e = routed ? blockIdx.z : 0;
    const float* Wg = Wg_base + (size_t)e * kI * kH;
    const float* Wu = Wu_base + (size_t)e * kI * kH;
    const int n_tok = routed ? cnt[e] : kT;
    const int m0 = blockIdx.y * kMT;
    if (m0 >= n_tok) return;
    const int ibase = blockIdx.x * 64;

    const int tid  = threadIdx.x;
    const int lane = tid & 31;
    const int wave = tid >> 5;
    const int half = lane >> 4;        // K half-select for A/B operands
    const int koff = half * 2;
    const int lm   = lane & 15;        // M (A) / N (B,C,D) index

    if (tid < kMT) {
        const int m = m0 + tid;
        int tok = 0; float s = 0.0f;
        if (m < n_tok) {
            tok = routed ? list[e * kT + m] : m;
            s   = routed ? sc[tok] : 1.0f;
        }
        tokLds[tid] = tok;
        sclLds[tid] = s;
    }

    const v8f vzero = {0.f, 0.f, 0.f, 0.f, 0.f, 0.f, 0.f, 0.f};
    v8f g[4], u[4];
#pragma unroll
    for (int ms = 0; ms < 4; ++ms) { g[ms] = vzero; u[ms] = vzero; }

    const int ncol = ibase + wave * 16 + lm;  // I column owned by this lane
    const float* wg_row = Wg + (size_t)ncol * kH;
    const float* wu_row = Wu + (size_t)ncol * kH;

    for (int kc = 0; kc < kH; kc += kKC) {
        // Prefetch next weight chunk into the cache hierarchy (streaming).
        if (kc + kKC < kH) {
            __builtin_prefetch(wg_row + kc + kKC, 0, 0);
            __builtin_prefetch(wu_row + kc + kKC, 0, 0);
        }
        __syncthreads();
        // Stage scaled/gathered A tile [64 x 32] into LDS (coalesced rows).
#pragma unroll
        for (int i = 0; i < (kMT * kKC) / 128; ++i) {
            const int idx = i * 128 + tid;
            const int m = idx / kKC;
            const int k = idx % kKC;
            float v = 0.0f;
            if (m0 + m < n_tok)
                v = X[(size_t)tokLds[m] * kH + kc + k] * sclLds[m];
            Alds[m * kLDA + k] = v;
        }
        __syncthreads();

        // Stage ALL B fragments of this chunk first: 16 clustered b64 loads,
        // one wait, then 64 back-to-back WMMAs.
        v2f bg[8], bu[8];
#pragma unroll
        for (int j = 0; j < 8; ++j) {
            bg[j] = *reinterpret_cast<const v2f*>(wg_row + kc + 4 * j + koff);
            bu[j] = *reinterpret_cast<const v2f*>(wu_row + kc + 4 * j + koff);
        }
#pragma unroll
        for (int j = 0; j < 8; ++j) {
            const int kk = 4 * j;
            v2f a[4];
#pragma unroll
            for (int ms = 0; ms < 4; ++ms) {
                const int ar = (ms * 16 + lm) * kLDA + kk + koff;
                a[ms] = v2f{Alds[ar], Alds[ar + 1]};
            }
#pragma unroll
            for (int ms = 0; ms < 4; ++ms) {
                g[ms] = wmma_f32(a[ms], bg[j], g[ms]);
                u[ms] = wmma_f32(a[ms], bu[j], u[ms]);
            }
        }
    }

    // Epilogue: P = silu(g)*u, scattered back to global token rows.
#pragma unroll
    for (int ms = 0; ms < 4; ++ms) {
#pragma unroll
        for (int r = 0; r < 8; ++r) {
            const int mloc = ms * 16 + half * 8 + r;
            if (m0 + mloc < n_tok) {
                const float pv = silu_f(g[ms][r]) * u[ms][r];
                P[(size_t)tokLds[mloc] * kI + ncol] = pv;
            }
        }
    }
}

// --------------------------- GEMM2: Y = P Wd^T -----------------------------
__global__ __launch_bounds__(128) void gemm2_kernel(
    const float* __restrict__ P,        // [T, I]
    const float* __restrict__ Wd_base,  // routed: [E,H,I]; shared: [H,I]
    float* __restrict__ out,            // [T, H]
    const int* __restrict__ cnt, const int* __restrict__ list,
    int routed) {                       // routed also selects += vs =
    __shared__ float Alds[kMT * kLDA];
    __shared__ int   tokLds[kMT];

    const int e = routed ? blockIdx.z : 0;
    const float* Wd = Wd_base + (size_t)e * kH * kI;
    const int n_tok = routed ? cnt[e] : kT;
    const int m0 = blockIdx.y * kMT;
    if (m0 >= n_tok) return;
    const int hbase = blockIdx.x * 64;

    const int tid  = threadIdx.x;
    const int lane = tid & 31;
    const int wave = tid >> 5;
    const int half = lane >> 4;
    const int koff = half * 2;
    const int lm   = lane & 15;

    if (tid < kMT) {
        const int m = m0 + tid;
        tokLds[tid] = (m < n_tok) ? (routed ? list[e * kT + m] : m) : 0;
    }

    const v8f vzero = {0.f, 0.f, 0.f, 0.f, 0.f, 0.f, 0.f, 0.f};
    v8f acc[4];
#pragma unroll
    for (int ms = 0; ms < 4; ++ms) acc[ms] = vzero;

    const int ncol = hbase + wave * 16 + lm;  // H column owned by this lane
    const float* wd_row = Wd + (size_t)ncol * kI;

    for (int kc = 0; kc < kI; kc += kKC) {
        if (kc + kKC < kI) {
            __builtin_prefetch(wd_row + kc + kKC, 0, 0);
        }
        __syncthreads();
#pragma unroll
        for (int i = 0; i < (kMT * kKC) / 128; ++i) {
            const int idx = i * 128 + tid;
            const int m = idx / kKC;
            const int k = idx % kKC;
            float v = 0.0f;
            if (m0 + m < n_tok)
                v = P[(size_t)tokLds[m] * kI + kc + k];
            Alds[m * kLDA + k] = v;
        }
        __syncthreads();

        v2f b[8];
#pragma unroll
        for (int j = 0; j < 8; ++j)
            b[j] = *reinterpret_cast<const v2f*>(wd_row + kc + 4 * j + koff);
#pragma unroll
        for (int j = 0; j < 8; ++j) {
            const int kk = 4 * j;
            v2f a[4];
#pragma unroll
            for (int ms = 0; ms < 4; ++ms) {
                const int ar = (ms * 16 + lm) * kLDA + kk + koff;
                a[ms] = v2f{Alds[ar], Alds[ar + 1]};
            }
#pragma unroll
            for (int ms = 0; ms < 4; ++ms)
                acc[ms] = wmma_f32(a[ms], b[j], acc[ms]);
        }
    }

#pragma unroll
    for (int ms = 0; ms < 4; ++ms) {
#pragma unroll
        for (int r = 0; r < 8; ++r) {
            const int mloc = ms * 16 + half * 8 + r;
            if (m0 + mloc < n_tok) {
                const size_t o = (size_t)tokLds[mloc] * kH + ncol;
                if (routed) out[o] += acc[ms][r];  // each token in exactly 1 list
                else        out[o]  = acc[ms][r];  // shared phase writes first
            }
        }
    }
}

// ------------------------------- launcher ----------------------------------
extern "C" void kernel_launch(void* const* d_in, const int* in_sizes, int n_in,
                              void* d_out, int out_size, void* d_ws, size_t ws_size,
                              hipStream_t stream) {
    (void)in_sizes; (void)n_in; (void)out_size; (void)ws_size;

    const float* hs      = (const float*)d_in[0];  // [1,T,H]
    const float* rw      = (const float*)d_in[1];  // [E,H]
    const float* w_gate  = (const float*)d_in[2];  // [E,I,H]
    const float* w_up    = (const float*)d_in[3];  // [E,I,H]
    const float* w_down  = (const float*)d_in[4];  // [E,H,I]
    const float* wsg     = (const float*)d_in[5];  // [I,H]
    const float* wsu     = (const float*)d_in[6];  // [I,H]
    const float* wsd     = (const float*)d_in[7];  // [H,I]

    float* out        = (float*)d_out;                     // [T,H]
    float* scores_out = out + (size_t)kT * kH;             // [E,T]

    char* wsp   = (char*)d_ws;
    float* P    = (float*)wsp;                                       // T*I f32
    float* sc   = (float*)(wsp + (size_t)kT * kI * sizeof(float));   // T f32
    int*   cnt  = (int*)((char*)sc + (size_t)kT * sizeof(float));    // E i32
    int*   list = (int*)((char*)cnt + kE * sizeof(int));             // E*T i32

    init_kernel<<<1, 32, 0, stream>>>(cnt);
    router_kernel<<<kT / 8, 256, 0, stream>>>(hs, rw, scores_out, sc, cnt, list);

    const dim3 blk(128);
    // Shared expert: writes out (covers every (t,h) once).
    gemm1_kernel<<<dim3(kI / 64, kT / kMT, 1), blk, 0, stream>>>(
        hs, wsg, wsu, P, cnt, list, sc, 0);
    gemm2_kernel<<<dim3(kH / 64, kT / kMT, 1), blk, 0, stream>>>(
        P, wsd, out, cnt, list, 0);
    // Routed experts (token-grouped; over-launched tiles early-exit).
    gemm1_kernel<<<dim3(kI / 64, kT / kMT, kE), blk, 0, stream>>>(
        hs, w_gate, w_up, P, cnt, list, sc, 1);
    gemm2_kernel<<<dim3(kH / 64, kT / kMT, kE), blk, 0, stream>>>(
        P, w_down, out, cnt, list, 1);
}